// WassersteinFeatureLoss_45217415692768
// MI455X (gfx1250) — compile-verified
//
#include <hip/hip_runtime.h>

// Problem constants (from reference)
#define N_CH   100      // channels used
#define N_SAMP 1000     // samples per channel
#define SORT_N 1024     // padded sort size (pow2)
#define CHANS  256
#define HW     3136     // 56*56
#define WAVE   32

// ---- LDS raw-offset helper: generic ptr -> addrspace(3) ptr -> 32-bit LDS byte addr
__device__ __forceinline__ unsigned lds_byte_addr(const void* p) {
    return (unsigned)(unsigned long long)
        (__attribute__((address_space(3))) const char*)p;
}

// ---- CDNA5 async global->LDS gather (per-lane global addr, per-lane LDS addr)
// GLOBAL_LOAD_ASYNC_TO_LDS_B32: per-lane dword copy global->LDS; tracked by ASYNCcnt
__device__ __forceinline__ void async_gather_b32(unsigned lds_addr,
                                                 unsigned long long gaddr) {
    asm volatile("global_load_async_to_lds_b32 %0, %1, off"
                 :: "v"(lds_addr), "v"(gaddr)
                 : "memory");
}

__device__ __forceinline__ void wait_async_all() {
    asm volatile("s_wait_asynccnt 0" ::: "memory");
}

// One workgroup (32 wave32 waves) per channel:
// async gather -> hybrid register/LDS bitonic sort of both arrays -> |diff| reduce.
__global__ __launch_bounds__(SORT_N)
void w1_per_channel(const float* __restrict__ sfeat,
                    const float* __restrict__ tfeat,
                    const int*   __restrict__ sidx,
                    float*       __restrict__ per_ch) {
    __shared__ float s_s[SORT_N];
    __shared__ float s_t[SORT_N];

    const unsigned tid = threadIdx.x;
    const unsigned ch  = blockIdx.x;

    if (tid < N_SAMP) {
        // flat index over (bs, h*w): f = b*HW + p ; element lives at ((b*C + ch)*HW + p)
        unsigned f = (unsigned)sidx[(size_t)ch * N_SAMP + tid];
        unsigned b = f / HW;
        unsigned p = f - b * HW;
        size_t off = ((size_t)b * CHANS + ch) * (size_t)HW + p;
        async_gather_b32(lds_byte_addr(&s_s[tid]),
                         (unsigned long long)(const void*)(sfeat + off));
        async_gather_b32(lds_byte_addr(&s_t[tid]),
                         (unsigned long long)(const void*)(tfeat + off));
    } else {
        // identical +INF padding sorts to the tail of both arrays and is skipped
        s_s[tid] = __builtin_inff();
        s_t[tid] = __builtin_inff();
    }
    wait_async_all();        // this wave's async LDS writes have landed
    __syncthreads();         // all waves' gathers visible

    // Register-resident elements: thread tid owns position tid of each array.
    float vs = s_s[tid];
    float vt = s_t[tid];

    // Bitonic sort. Steps with j<32 are intra-wave -> shuffle exchange, no barrier.
    // Steps with j>=32 cross waves -> LDS exchange (15 of 55 steps).
    #pragma unroll 1
    for (unsigned k = 2; k <= SORT_N; k <<= 1) {
        #pragma unroll 1
        for (unsigned j = k >> 1; j > 0; j >>= 1) {
            // ascending region iff (tid&k)==0; lower partner keeps min when ascending
            const bool keep_min = ((tid & k) == 0) == ((tid & j) == 0);
            float os, ot;
            if (j >= WAVE) {
                __syncthreads();          // protect LDS from previous iteration reads
                s_s[tid] = vs;
                s_t[tid] = vt;
                __syncthreads();
                os = s_s[tid ^ j];
                ot = s_t[tid ^ j];
            } else {
                os = __shfl_xor(vs, (int)j, WAVE);
                ot = __shfl_xor(vt, (int)j, WAVE);
            }
            vs = keep_min ? fminf(vs, os) : fmaxf(vs, os);
            vt = keep_min ? fminf(vt, ot) : fmaxf(vt, ot);
        }
    }

    // W1 contribution: |sorted_s - sorted_t| over the valid 1000 positions.
    float d = (tid < N_SAMP) ? fabsf(vs - vt) : 0.0f;
    // wave32 shuffle reduction
    #pragma unroll
    for (int o = WAVE / 2; o > 0; o >>= 1) d += __shfl_xor(d, o, WAVE);
    // cross-wave: 32 partial sums through LDS (s_s safe to reuse: last LDS read
    // in the sort is followed by shuffle-only steps)
    __syncthreads();
    if ((tid & (WAVE - 1)) == 0) s_s[tid >> 5] = d;
    __syncthreads();
    if (tid < WAVE) {
        float x = s_s[tid];
        #pragma unroll
        for (int o = WAVE / 2; o > 0; o >>= 1) x += __shfl_xor(x, o, WAVE);
        if (tid == 0) per_ch[ch] = x * (1.0f / (float)N_SAMP);
    }
}

// Deterministic final mean over the 100 per-channel W1 values.
__global__ __launch_bounds__(128)
void w1_finalize(const float* __restrict__ per_ch, float* __restrict__ out) {
    __shared__ float red[128];
    const unsigned tid = threadIdx.x;
    red[tid] = (tid < N_CH) ? per_ch[tid] : 0.0f;
    __syncthreads();
    #pragma unroll 1
    for (unsigned s = 64; s > 0; s >>= 1) {
        if (tid < s) red[tid] += red[tid + s];
        __syncthreads();
    }
    if (tid == 0) out[0] = red[0] * (1.0f / (float)N_CH);
}

extern "C" void kernel_launch(void* const* d_in, const int* in_sizes, int n_in,
                              void* d_out, int out_size, void* d_ws, size_t ws_size,
                              hipStream_t stream) {
    const float* sfeat = (const float*)d_in[0];
    const float* tfeat = (const float*)d_in[1];
    const int*   sidx  = (const int*)d_in[2];   // harness convention: integer -> int32
    float* per_ch = (float*)d_ws;               // 100 floats of scratch
    float* out    = (float*)d_out;

    w1_per_channel<<<N_CH, SORT_N, 0, stream>>>(sfeat, tfeat, sidx, per_ch);
    w1_finalize<<<1, 128, 0, stream>>>(per_ch, out);
}